// TUBA_28724741276086
// MI455X (gfx1250) — compile-verified
//
#include <hip/hip_runtime.h>
#include <hip/hip_bf16.h>

// ---------------------------------------------------------------------------
// Problem constants (fixed by the reference): N=1024, X_DIM=Y_DIM=128, HID=256
// ---------------------------------------------------------------------------
#define XD   128
#define HID  256

typedef float v2f __attribute__((ext_vector_type(2)));
typedef float v8f __attribute__((ext_vector_type(8)));

// ---------------------------------------------------------------------------
// Kernel 1: fp32 WMMA projection.
//   z==0:  aOut  = x @ Wx^T + b1        (Wx = W1[:, 0:128])
//   z==1:  pyOut = y @ Wy^T             (Wy = W1[:, 128:256])
// One wave (32 lanes) per 16x16 output tile; K-loop of V_WMMA_F32_16X16X4_F32.
// A-layout (16x4 f32): lanes 0-15 hold M=lane, VGPR0=K0,VGPR1=K1;
//                      lanes 16-31 hold M=lane-16, VGPR0=K2,VGPR1=K3.
// B-layout (4x16 f32): symmetric with N in place of M.
// C/D (16x16 f32, 8 VGPRs): lane n=(l&15), rows M = 8*(l>>4) + v.
// ---------------------------------------------------------------------------
__global__ __launch_bounds__(32) void tuba_proj_wmma(
    const float* __restrict__ x, const float* __restrict__ y,
    const float* __restrict__ W1, const float* __restrict__ b1,
    float* __restrict__ aOut, float* __restrict__ pyOut)
{
    const int lane = threadIdx.x & 31;
    const int mr   = lane & 15;    // M row (for A) / N col (for B)
    const int kh   = lane >> 4;    // which K-half this lane carries
    const int row0 = blockIdx.x * 16;
    const int col0 = blockIdx.y * 16;
    const bool isX = (blockIdx.z == 0);

    const float* __restrict__ src  = isX ? x : y;
    const int    woff              = isX ? 0 : XD;
    const float* __restrict__ arow = src + (size_t)(row0 + mr) * XD;
    const float* __restrict__ wrow = W1  + (size_t)(col0 + mr) * (2 * XD) + woff;

    v8f c = {0.f, 0.f, 0.f, 0.f, 0.f, 0.f, 0.f, 0.f};
#pragma unroll
    for (int k = 0; k < XD; k += 4) {
        const int kk = k + 2 * kh;
        v2f av; av.x = arow[kk]; av.y = arow[kk + 1];
        v2f bv; bv.x = wrow[kk]; bv.y = wrow[kk + 1];
        c = __builtin_amdgcn_wmma_f32_16x16x4_f32(
                /*neg_a=*/false, av, /*neg_b=*/false, bv,
                /*c_mod=*/(short)0, c, /*reuse_a=*/false, /*reuse_b=*/false);
    }

    const int n  = lane & 15;
    const int mb = kh * 8;
    float* __restrict__ dst = isX ? aOut : pyOut;
#pragma unroll
    for (int v = 0; v < 8; ++v) {
        float val = c[v];
        const int cc = col0 + n;
        if (isX) val += b1[cc];                    // fold b1 into the x-projection
        dst[(size_t)(row0 + mb + v) * HID + cc] = val;
    }
}

// ---------------------------------------------------------------------------
// Kernel 2: log_scores[r] = y[r,:] . wb + bb   (trivial GEMV, fully L2-resident)
// ---------------------------------------------------------------------------
__global__ __launch_bounds__(256) void tuba_logscore(
    const float* __restrict__ y, const float* __restrict__ wb,
    const float* __restrict__ bbp, float* __restrict__ logsc, int n)
{
    const int r = blockIdx.x * blockDim.x + threadIdx.x;
    if (r >= n) return;
    float s = *bbp;
    const float* __restrict__ row = y + (size_t)r * XD;
#pragma unroll 4
    for (int d = 0; d < XD; ++d) s = fmaf(row[d], wb[d], s);
    logsc[r] = s;
}

// ---------------------------------------------------------------------------
// Kernel 3: pairwise VALU core.  64x64 (i,j) tile / block, 256 threads,
// 4x4 register tile per thread, k-chunked transposed LDS (ds_load_b128 feeds).
//   T1[i,j] = sum_k relu(a[j,k] + py[i,k]) * w2[k] + b2
//   S3 += exp(T1[i,j] - log_scores[j]);  S1 += T1[j,j] on the diagonal.
// ---------------------------------------------------------------------------
#define TI 64
#define TJ 64
#define KC 64

__global__ __launch_bounds__(256) void tuba_pair(
    const float* __restrict__ aBuf,   // (n, HID) = px + b1
    const float* __restrict__ pyBuf,  // (n, HID)
    const float* __restrict__ w2,
    const float* __restrict__ b2p,
    const float* __restrict__ logsc,
    float* __restrict__ accum)        // accum[0] = S3, accum[1] = S1
{
    __shared__ __align__(16) float aT[KC][TJ + 4];   // transposed: [k][j]
    __shared__ __align__(16) float pT[KC][TI + 4];   // transposed: [k][i]
    __shared__ float w2s[HID];
    __shared__ float lscS[TJ];
    __shared__ float red[256];

    const int tid = threadIdx.x;
    const int j0  = blockIdx.x * TJ;
    const int i0  = blockIdx.y * TI;
    const int tx  = tid & 15;
    const int ty  = tid >> 4;
    const int jl  = tx * 4;
    const int il  = ty * 4;

    w2s[tid] = w2[tid & (HID - 1)];
    if (tid >= 256 - TJ) { /* nothing */ }
    if (tid < TJ) lscS[tid] = logsc[j0 + tid];

    float acc[4][4];
#pragma unroll
    for (int a = 0; a < 4; ++a)
#pragma unroll
        for (int b = 0; b < 4; ++b) acc[a][b] = 0.f;

    for (int kc = 0; kc < HID; kc += KC) {
        __syncthreads();
        // Stage one 64-row x 64-k chunk of each matrix, transposed into LDS.
        // 1024 float4 loads per matrix -> 4 iterations of 256 threads.
#pragma unroll
        for (int it = 0; it < 4; ++it) {
            const int idx = it * 256 + tid;
            const int row = idx >> 4;          // 0..63
            const int k4  = (idx & 15) * 4;    // 0..60
            const float4 va = *(const float4*)(aBuf  + (size_t)(j0 + row) * HID + kc + k4);
            aT[k4 + 0][row] = va.x; aT[k4 + 1][row] = va.y;
            aT[k4 + 2][row] = va.z; aT[k4 + 3][row] = va.w;
            const float4 vp = *(const float4*)(pyBuf + (size_t)(i0 + row) * HID + kc + k4);
            pT[k4 + 0][row] = vp.x; pT[k4 + 1][row] = vp.y;
            pT[k4 + 2][row] = vp.z; pT[k4 + 3][row] = vp.w;
        }
        __syncthreads();

#pragma unroll 4
        for (int k = 0; k < KC; ++k) {
            const float4 av = *(const float4*)&aT[k][jl];
            const float4 pv = *(const float4*)&pT[k][il];
            const float  wk = w2s[kc + k];
            const float aj[4] = {av.x, av.y, av.z, av.w};
            const float pi[4] = {pv.x, pv.y, pv.z, pv.w};
#pragma unroll
            for (int a = 0; a < 4; ++a)
#pragma unroll
                for (int b = 0; b < 4; ++b) {
                    float t = pi[a] + aj[b];
                    t = fmaxf(t, 0.f);
                    acc[a][b] = fmaf(t, wk, acc[a][b]);
                }
        }
    }

    // Epilogue: exp-sum (S3) and diagonal T1 (S1).
    const float b2v = *b2p;
    float s3 = 0.f, s1 = 0.f;
#pragma unroll
    for (int a = 0; a < 4; ++a) {
        const int gi = i0 + il + a;
#pragma unroll
        for (int b = 0; b < 4; ++b) {
            const int gj = j0 + jl + b;
            const float t1 = acc[a][b] + b2v;
            s3 += __expf(t1 - lscS[jl + b]);
            if (gi == gj) s1 += t1;
        }
    }

    // Block reduction, then one atomic per block per sum.
    red[tid] = s3; __syncthreads();
    for (int s = 128; s > 0; s >>= 1) {
        if (tid < s) red[tid] += red[tid + s];
        __syncthreads();
    }
    if (tid == 0) atomicAdd(&accum[0], red[0]);
    __syncthreads();
    red[tid] = s1; __syncthreads();
    for (int s = 128; s > 0; s >>= 1) {
        if (tid < s) red[tid] += red[tid + s];
        __syncthreads();
    }
    if (tid == 0) atomicAdd(&accum[1], red[0]);
}

// ---------------------------------------------------------------------------
// Kernel 4: finalize.  S2 = sum(log_scores); emit the scalar bound.
// ---------------------------------------------------------------------------
__global__ __launch_bounds__(256) void tuba_finalize(
    const float* __restrict__ logsc, const float* __restrict__ accum,
    float* __restrict__ out, int n)
{
    __shared__ float red[256];
    float s = 0.f;
    for (int r = threadIdx.x; r < n; r += 256) s += logsc[r];
    red[threadIdx.x] = s; __syncthreads();
    for (int st = 128; st > 0; st >>= 1) {
        if (threadIdx.x < st) red[threadIdx.x] += red[threadIdx.x + st];
        __syncthreads();
    }
    if (threadIdx.x == 0) {
        const float fn = (float)n;
        // 1 + mean(T0) - mean(log_scores) - mean_i( sum_j exp(T1-ls) / n )
        out[0] = 1.0f + accum[1] / fn - red[0] / fn - accum[0] / (fn * fn);
    }
}

// ---------------------------------------------------------------------------
// Launcher
// ---------------------------------------------------------------------------
extern "C" void kernel_launch(void* const* d_in, const int* in_sizes, int n_in,
                              void* d_out, int out_size, void* d_ws, size_t ws_size,
                              hipStream_t stream) {
    const float* x  = (const float*)d_in[0];
    const float* y  = (const float*)d_in[1];
    const float* W1 = (const float*)d_in[2];
    const float* b1 = (const float*)d_in[3];
    const float* w2 = (const float*)d_in[4];
    const float* b2 = (const float*)d_in[5];
    const float* wb = (const float*)d_in[6];
    const float* bb = (const float*)d_in[7];
    float* out = (float*)d_out;

    const int n = in_sizes[0] / XD;   // 1024

    float* aBuf  = (float*)d_ws;                 // (n, HID)  px + b1
    float* pyBuf = aBuf  + (size_t)n * HID;      // (n, HID)
    float* logsc = pyBuf + (size_t)n * HID;      // (n,)
    float* accum = logsc + n;                    // [S3, S1]

    hipMemsetAsync(accum, 0, 2 * sizeof(float), stream);

    tuba_proj_wmma<<<dim3(n / 16, HID / 16, 2), 32, 0, stream>>>(
        x, y, W1, b1, aBuf, pyBuf);

    tuba_logscore<<<dim3((n + 255) / 256), 256, 0, stream>>>(y, wb, bb, logsc, n);

    tuba_pair<<<dim3(n / TJ, n / TI), 256, 0, stream>>>(
        aBuf, pyBuf, w2, b2, logsc, accum);

    tuba_finalize<<<1, 256, 0, stream>>>(logsc, accum, out, n);
}